// GATDecoder_18605798326838
// MI455X (gfx1250) — compile-verified
//
#include <hip/hip_runtime.h>
#include <cstdint>

#define N_NODES   50000
#define N_EDGES   800000
#define GAT_SLOPE 0.2f
#define ACT_SLOPE 0.01f
#define BN_EPS    1e-5f

typedef __attribute__((ext_vector_type(2))) float v2f;
typedef __attribute__((ext_vector_type(8))) float v8f;
typedef __attribute__((ext_vector_type(4))) int   v4i;

typedef __attribute__((address_space(1))) v4i* glb_v4i_ptr;
typedef __attribute__((address_space(3))) v4i* lds_v4i_ptr;

#if defined(__has_builtin)
#if __has_builtin(__builtin_amdgcn_global_load_async_to_lds_b128)
#define HAVE_ASYNC_LDS_B128 1
#endif
#endif

__device__ __forceinline__ float leaky(float x, float s) { return x > 0.f ? x : s * x; }

// float atomic max via integer ordering trick (monotone for IEEE bits)
__device__ __forceinline__ void atomicMaxF(float* addr, float val) {
    if (val >= 0.f) atomicMax((int*)addr, __float_as_int(val));
    else            atomicMin((unsigned int*)addr, __float_as_uint(val));
}

#if defined(HAVE_ASYNC_LDS_B128)
__device__ __forceinline__ void wait_async0() {
#if __has_builtin(__builtin_amdgcn_s_wait_asynccnt)
    __builtin_amdgcn_s_wait_asynccnt(0);
#else
    asm volatile("s_wait_asynccnt 0x0" ::: "memory");
#endif
}
#endif

// ---------------------------------------------------------------------------
// fill
// ---------------------------------------------------------------------------
__global__ void fill_kernel(float* __restrict__ p, float v, long n) {
    long i = blockIdx.x * (long)blockDim.x + threadIdx.x;
    long s = (long)gridDim.x * blockDim.x;
    for (; i < n; i += s) p[i] = v;
}

// ---------------------------------------------------------------------------
// fp32 WMMA GEMM: Y[nrows x OUT] = X[nrows x IN] @ W[IN x OUT] (+ bias)
// 8 waves per block; wave w owns 16-row M-tile (blockIdx.x*8 + w).
// A fragment (16 x IN) hoisted to registers once; W slab (IN x 16) staged to
// LDS with GLOBAL_LOAD_ASYNC_TO_LDS_B128 (ASYNCcnt) when available.
// V_WMMA_F32_16X16X4_F32: A 16x4, B 4x16 per step (ISA 7.12.2 layouts).
// ---------------------------------------------------------------------------
template <int IN, int OUT, bool BIAS>
__global__ __launch_bounds__(256) void wmma_gemm(const float* __restrict__ X,
                                                 const float* __restrict__ W,
                                                 const float* __restrict__ bias,
                                                 float* __restrict__ Y, int nrows) {
    __shared__ __align__(16) float Ws[IN * 16];
    const int tid  = threadIdx.x;
    const int wave = tid >> 5;
    const int lane = tid & 31;
    const int row16 = (blockIdx.x * 8 + wave) * 16;   // first row of this wave's tile
    const int mr   = lane & 15;                       // A-matrix row within tile
    const int koff = (lane >> 4) * 2;                 // lanes 16-31 supply K+2,K+3
    const int n    = lane & 15;                       // B/D column within tile
    const int hi   = (lane >> 4) * 8;                 // D rows 8..15 in upper half

    // Hoist A fragments: lane holds X[row16+mr][k*4+koff .. +1] for all k
    float a0[IN / 4], a1[IN / 4];
    {
        const int row = row16 + mr;
        const bool ok = row < nrows;
        const float* xr = X + (long)row * IN;
#pragma unroll
        for (int k = 0; k < IN / 4; ++k) {
            a0[k] = ok ? xr[k * 4 + koff]     : 0.f;
            a1[k] = ok ? xr[k * 4 + koff + 1] : 0.f;
        }
    }

    for (int nt = 0; nt < OUT / 16; ++nt) {
        __syncthreads();   // previous slab consumers done
#if defined(HAVE_ASYNC_LDS_B128)
        // direct memory->LDS DMA, tracked by ASYNCcnt (no VGPR round trip)
        for (int idx = tid * 4; idx < IN * 16; idx += 256 * 4) {
            const int k = idx >> 4, nn = idx & 15;
            __builtin_amdgcn_global_load_async_to_lds_b128(
                (glb_v4i_ptr)(W + (long)k * OUT + nt * 16 + nn),
                (lds_v4i_ptr)(&Ws[idx]),
                0, 0);
            if (nt + 1 < OUT / 16)
                __builtin_prefetch(&W[(long)k * OUT + (nt + 1) * 16 + nn], 0, 1);
        }
        wait_async0();
#else
        for (int idx = tid; idx < IN * 16; idx += 256) {
            Ws[idx] = W[(idx >> 4) * OUT + nt * 16 + (idx & 15)];
            if (nt + 1 < OUT / 16 && (idx & 31) == 0)   // one prefetch per 128B line
                __builtin_prefetch(&W[(idx >> 4) * OUT + (nt + 1) * 16], 0, 1);
        }
#endif
        __syncthreads();

        v8f acc = {};
#pragma unroll
        for (int k = 0; k < IN / 4; ++k) {
            v2f a; a[0] = a0[k]; a[1] = a1[k];
            v2f b;
            b[0] = Ws[(k * 4 + koff) * 16 + n];
            b[1] = Ws[(k * 4 + koff + 1) * 16 + n];
            acc = __builtin_amdgcn_wmma_f32_16x16x4_f32(
                false, a, false, b, (short)0, acc, false, false);
        }

#pragma unroll
        for (int r = 0; r < 8; ++r) {
            const int row = row16 + r + hi;           // C/D layout: M = r + 8*(lane>=16)
            if (row < nrows) {
                float v = acc[r];
                if (BIAS) v += bias[nt * 16 + n];
                Y[(long)row * OUT + nt * 16 + n] = v;
            }
        }
    }
}

// ---------------------------------------------------------------------------
// per-node attention scalars: e = H.a_src, f = H.a_dst  (one wave per node)
// ---------------------------------------------------------------------------
template <int C>
__global__ __launch_bounds__(256) void ef_kernel(const float* __restrict__ H,
                                                 const float* __restrict__ as,
                                                 const float* __restrict__ ad,
                                                 float* __restrict__ e, float* __restrict__ f,
                                                 int nrows) {
    const int wave = threadIdx.x >> 5, lane = threadIdx.x & 31;
    const int node = blockIdx.x * 8 + wave;
    if (node >= nrows) return;
    float se = 0.f, sf = 0.f;
    for (int c = lane; c < C; c += 32) {
        const float h = H[(long)node * C + c];
        se += h * as[c];
        sf += h * ad[c];
    }
#pragma unroll
    for (int off = 16; off; off >>= 1) {
        se += __shfl_down(se, off, 32);
        sf += __shfl_down(sf, off, 32);
    }
    if (lane == 0) { e[node] = se; f[node] = sf; }
}

// ---------------------------------------------------------------------------
// edge passes (E real edges + N implicit self-loops appended, like reference)
// ---------------------------------------------------------------------------
__device__ __forceinline__ void edge_sd(const long long* __restrict__ ei, int t, int nE,
                                        int& s, int& d) {
    if (t < nE) { s = (int)ei[t]; d = (int)ei[nE + t]; }
    else        { s = d = t - nE; }
}

__global__ void edge_max_kernel(const long long* __restrict__ ei,
                                const float* __restrict__ e, const float* __restrict__ f,
                                float* __restrict__ m, int nE, int nN) {
    const int total = nE + nN;
    for (int t = blockIdx.x * blockDim.x + threadIdx.x; t < total;
         t += gridDim.x * blockDim.x) {
        int s, d; edge_sd(ei, t, nE, s, d);
        atomicMaxF(&m[d], leaky(e[s] + f[d], GAT_SLOPE));
    }
}

__global__ void edge_exp_kernel(const long long* __restrict__ ei,
                                const float* __restrict__ e, const float* __restrict__ f,
                                const float* __restrict__ m, float* __restrict__ denom,
                                float* __restrict__ ex, int nE, int nN) {
    const int total = nE + nN;
    for (int t = blockIdx.x * blockDim.x + threadIdx.x; t < total;
         t += gridDim.x * blockDim.x) {
        int s, d; edge_sd(ei, t, nE, s, d);
        const float lg = leaky(e[s] + f[d], GAT_SLOPE);
        const float v  = __expf(lg - m[d]);
        ex[t] = v;
        unsafeAtomicAdd(&denom[d], v);
    }
}

template <int C>
__global__ void edge_aggregate_kernel(const long long* __restrict__ ei,
                                      const float* __restrict__ H,
                                      const float* __restrict__ ex,
                                      const float* __restrict__ denom,
                                      float* __restrict__ agg, int nE, int nN) {
    const long total = (long)(nE + nN) * (C / 4);
    for (long t = blockIdx.x * (long)blockDim.x + threadIdx.x; t < total;
         t += (long)gridDim.x * blockDim.x) {
        const int eidx = (int)(t / (C / 4));
        const int c4   = (int)(t % (C / 4)) * 4;
        int s, d; edge_sd(ei, eidx, nE, s, d);
        const float alpha = ex[eidx] / denom[d];
        const float4 h = *reinterpret_cast<const float4*>(&H[(long)s * C + c4]);
        float* o = &agg[(long)d * C + c4];
        unsafeAtomicAdd(o + 0, alpha * h.x);
        unsafeAtomicAdd(o + 1, alpha * h.y);
        unsafeAtomicAdd(o + 2, alpha * h.z);
        unsafeAtomicAdd(o + 3, alpha * h.w);
    }
}

// ---------------------------------------------------------------------------
// BatchNorm: per-feature sum/sumsq (block-local reduce + 1 atomic per block),
// then fused normalize + LeakyReLU(0.01) in place.
// ---------------------------------------------------------------------------
template <int C>
__global__ __launch_bounds__(256) void bn_stats_kernel(const float* __restrict__ X,
                                                       float* __restrict__ sum,
                                                       float* __restrict__ sumsq, int nrows) {
    const int c  = threadIdx.x;       // blockDim.x == C
    const int r0 = blockIdx.x * 256;
    int r1 = r0 + 256; if (r1 > nrows) r1 = nrows;
    float s = 0.f, sq = 0.f;
    for (int r = r0; r < r1; ++r) {
        const float x = X[(long)r * C + c];
        s += x; sq += x * x;
    }
    unsafeAtomicAdd(&sum[c], s);
    unsafeAtomicAdd(&sumsq[c], sq);
}

template <int C>
__global__ void bn_apply_kernel(float* __restrict__ X, const float* __restrict__ sum,
                                const float* __restrict__ sumsq,
                                const float* __restrict__ gamma,
                                const float* __restrict__ beta, int nrows) {
    const long total = (long)nrows * C;
    const float inv = 1.f / (float)nrows;
    for (long t = blockIdx.x * (long)blockDim.x + threadIdx.x; t < total;
         t += (long)gridDim.x * blockDim.x) {
        const int c = (int)(t % C);
        const float mean = sum[c] * inv;
        const float var  = sumsq[c] * inv - mean * mean;
        const float y = (X[t] - mean) * rsqrtf(var + BN_EPS) * gamma[c] + beta[c];
        X[t] = leaky(y, ACT_SLOPE);
    }
}

// ---------------------------------------------------------------------------
// launch
// ---------------------------------------------------------------------------
extern "C" void kernel_launch(void* const* d_in, const int* in_sizes, int n_in,
                              void* d_out, int out_size, void* d_ws, size_t ws_size,
                              hipStream_t stream) {
    const float*     emb = (const float*)d_in[0];
    const long long* ei  = (const long long*)d_in[1];   // jnp.int64 [2, E]
    const float* W1  = (const float*)d_in[2];
    const float* as1 = (const float*)d_in[3];
    const float* ad1 = (const float*)d_in[4];
    // d_in[5] = b1: cancels exactly under BatchNorm -> skipped
    const float* g1  = (const float*)d_in[6];
    const float* be1 = (const float*)d_in[7];
    const float* W2  = (const float*)d_in[8];
    const float* as2 = (const float*)d_in[9];
    const float* ad2 = (const float*)d_in[10];
    // d_in[11] = b2: cancels under BatchNorm -> skipped
    const float* g2  = (const float*)d_in[12];
    const float* be2 = (const float*)d_in[13];
    const float* Wf  = (const float*)d_in[14];
    const float* bf  = (const float*)d_in[15];
    float* out = (float*)d_out;

    // workspace layout (floats)
    float* ws = (float*)d_ws;
    long o = 0;
    float* H1   = ws + o; o += (long)N_NODES * 128;
    float* agg1 = ws + o; o += (long)N_NODES * 128;
    float* H2   = ws + o; o += (long)N_NODES * 256;
    float* agg2 = ws + o; o += (long)N_NODES * 256;
    float* ev   = ws + o; o += N_NODES;
    float* fv   = ws + o; o += N_NODES;
    float* mv   = ws + o; o += N_NODES;
    float* dv   = ws + o; o += N_NODES;
    float* exb  = ws + o; o += (long)N_EDGES + N_NODES;
    float* bsum = ws + o; o += 256;
    float* bsq  = ws + o; o += 256;   // bsum..bsq contiguous (512 floats)

    const dim3 blk(256);
    const int gemmGrid = ((N_NODES / 16) + 7) / 8;   // 3125 M-tiles, 8 waves/block
    const int efGrid   = (N_NODES + 7) / 8;
    const int bnGrid   = (N_NODES + 255) / 256;

    // ---------------- layer 1 (64 -> 128) ----------------
    fill_kernel<<<256, blk, 0, stream>>>(mv, -1e30f, N_NODES);
    fill_kernel<<<256, blk, 0, stream>>>(dv, 0.f, N_NODES);
    fill_kernel<<<2048, blk, 0, stream>>>(agg1, 0.f, (long)N_NODES * 128);
    fill_kernel<<<1, blk, 0, stream>>>(bsum, 0.f, 512);

    wmma_gemm<64, 128, false><<<gemmGrid, blk, 0, stream>>>(emb, W1, nullptr, H1, N_NODES);
    ef_kernel<128><<<efGrid, blk, 0, stream>>>(H1, as1, ad1, ev, fv, N_NODES);
    edge_max_kernel<<<2048, blk, 0, stream>>>(ei, ev, fv, mv, N_EDGES, N_NODES);
    edge_exp_kernel<<<2048, blk, 0, stream>>>(ei, ev, fv, mv, dv, exb, N_EDGES, N_NODES);
    edge_aggregate_kernel<128><<<4096, blk, 0, stream>>>(ei, H1, exb, dv, agg1, N_EDGES, N_NODES);
    bn_stats_kernel<128><<<bnGrid, 128, 0, stream>>>(agg1, bsum, bsq, N_NODES);
    bn_apply_kernel<128><<<2048, blk, 0, stream>>>(agg1, bsum, bsq, g1, be1, N_NODES);

    // ---------------- layer 2 (128 -> 256) ----------------
    fill_kernel<<<256, blk, 0, stream>>>(mv, -1e30f, N_NODES);
    fill_kernel<<<256, blk, 0, stream>>>(dv, 0.f, N_NODES);
    fill_kernel<<<2048, blk, 0, stream>>>(agg2, 0.f, (long)N_NODES * 256);
    fill_kernel<<<1, blk, 0, stream>>>(bsum, 0.f, 512);

    wmma_gemm<128, 256, false><<<gemmGrid, blk, 0, stream>>>(agg1, W2, nullptr, H2, N_NODES);
    ef_kernel<256><<<efGrid, blk, 0, stream>>>(H2, as2, ad2, ev, fv, N_NODES);
    edge_max_kernel<<<2048, blk, 0, stream>>>(ei, ev, fv, mv, N_EDGES, N_NODES);
    edge_exp_kernel<<<2048, blk, 0, stream>>>(ei, ev, fv, mv, dv, exb, N_EDGES, N_NODES);
    edge_aggregate_kernel<256><<<4096, blk, 0, stream>>>(ei, H2, exb, dv, agg2, N_EDGES, N_NODES);
    bn_stats_kernel<256><<<bnGrid, 256, 0, stream>>>(agg2, bsum, bsq, N_NODES);
    bn_apply_kernel<256><<<2048, blk, 0, stream>>>(agg2, bsum, bsq, g2, be2, N_NODES);

    // ---------------- final linear (256 -> 256), bias fused ----------------
    wmma_gemm<256, 256, true><<<gemmGrid, blk, 0, stream>>>(agg2, Wf, bf, out, N_NODES);
}